// FeaturePropagation_17824114278741
// MI455X (gfx1250) — compile-verified
//
#include <hip/hip_runtime.h>

#define B_      4
#define NC_     4096
#define NF_     16384
#define CC_     128
#define IN_CH_  256
#define OUT_CH_ 128
#define EPS_    1e-5f

typedef float v2f __attribute__((ext_vector_type(2)));
typedef float v8f __attribute__((ext_vector_type(8)));

// ---------------------------------------------------------------------------
// Kernel 1: brute-force 3-NN per fine point. Coarse xyz (48KB) cached in LDS.
// ---------------------------------------------------------------------------
__global__ __launch_bounds__(256)
void knn_kernel(const float* __restrict__ xyz_fine,
                const float* __restrict__ xyz_coarse,
                int* __restrict__ idx_out,
                float* __restrict__ w_out) {
    __shared__ float sc[NC_ * 3];
    const int blocksPerBatch = NF_ / 256;
    const int b = blockIdx.x / blocksPerBatch;
    const int n = (blockIdx.x % blocksPerBatch) * 256 + threadIdx.x;

    for (int i = threadIdx.x; i < NC_ * 3; i += 256)
        sc[i] = xyz_coarse[(size_t)b * NC_ * 3 + i];
    __syncthreads();

    const int p = b * NF_ + n;
    const float fx = xyz_fine[(size_t)p * 3 + 0];
    const float fy = xyz_fine[(size_t)p * 3 + 1];
    const float fz = xyz_fine[(size_t)p * 3 + 2];

    float d0 = 3.4e38f, d1 = 3.4e38f, d2 = 3.4e38f;
    int   i0 = 0, i1 = 0, i2 = 0;
    for (int j = 0; j < NC_; ++j) {
        const float dx = fx - sc[3 * j + 0];
        const float dy = fy - sc[3 * j + 1];
        const float dz = fz - sc[3 * j + 2];
        const float d  = fmaf(dx, dx, fmaf(dy, dy, dz * dz));
        if (d < d2) {
            if (d < d1) {
                if (d < d0) { d2 = d1; i2 = i1; d1 = d0; i1 = i0; d0 = d; i0 = j; }
                else        { d2 = d1; i2 = i1; d1 = d;  i1 = j; }
            } else          { d2 = d;  i2 = j; }
        }
    }

    d0 = sqrtf(fmaxf(d0, 0.f));
    d1 = sqrtf(fmaxf(d1, 0.f));
    d2 = sqrtf(fmaxf(d2, 0.f));

    float w0, w1, w2;
    if (d0 <= 1e-12f) {          // zero-distance: one-hot on nearest (index 0, sorted)
        w0 = 1.f; w1 = 0.f; w2 = 0.f;
    } else {
        w0 = 1.f / (d0 + 1e-12f);
        w1 = 1.f / (d1 + 1e-12f);
        w2 = 1.f / (d2 + 1e-12f);
        const float s = 1.f / (w0 + w1 + w2);
        w0 *= s; w1 *= s; w2 *= s;
    }
    idx_out[(size_t)p * 3 + 0] = i0;
    idx_out[(size_t)p * 3 + 1] = i1;
    idx_out[(size_t)p * 3 + 2] = i2;
    w_out[(size_t)p * 3 + 0] = w0;
    w_out[(size_t)p * 3 + 1] = w1;
    w_out[(size_t)p * 3 + 2] = w2;
}

// ---------------------------------------------------------------------------
// Kernel 2: weighted gather of 3 coarse-feature rows + concat skip features.
// One block per fine point, 128 threads = channels (coalesced rows).
// ---------------------------------------------------------------------------
__global__ __launch_bounds__(128)
void interp_concat_kernel(const float* __restrict__ feat_coarse,
                          const float* __restrict__ feat_skip,
                          const int* __restrict__ idx,
                          const float* __restrict__ w,
                          float* __restrict__ feat) {
    const int row = blockIdx.x;          // 0 .. B*NF-1
    const int b   = row / NF_;
    const int c   = threadIdx.x;         // 0..127
    const int   i0 = idx[(size_t)row * 3 + 0];
    const int   i1 = idx[(size_t)row * 3 + 1];
    const int   i2 = idx[(size_t)row * 3 + 2];
    const float w0 = w[(size_t)row * 3 + 0];
    const float w1 = w[(size_t)row * 3 + 1];
    const float w2 = w[(size_t)row * 3 + 2];

    const size_t base = (size_t)b * NC_ * CC_;
    float v = w0 * feat_coarse[base + (size_t)i0 * CC_ + c];
    v = fmaf(w1, feat_coarse[base + (size_t)i1 * CC_ + c], v);
    v = fmaf(w2, feat_coarse[base + (size_t)i2 * CC_ + c], v);

    feat[(size_t)row * IN_CH_ + c]       = v;
    feat[(size_t)row * IN_CH_ + CC_ + c] = feat_skip[(size_t)row * CC_ + c];
}

// ---------------------------------------------------------------------------
// Kernel 3: C[M,128] = A[M,K] @ W[K,128] + bias, fp32 WMMA 16x16x4.
// Block = 256 threads (8 waves), M-tile = 128 rows, K staged in 64-deep LDS
// chunks. A rows padded to 66 floats (even -> 8B-aligned ds_load_b64 for the
// {k,k+1} fragment pair; stride 66 -> conflict-free banks). W is staged
// *fragment-major*: each B fragment's two values (rows k0+half, k0+half+2)
// stored adjacently as a float2 so the B fragment is a single ds_load_b64
// straight into the even-aligned VGPR pair WMMA wants (no repack moves).
// ---------------------------------------------------------------------------
__global__ __launch_bounds__(256)
void gemm_bias_wmma(const float* __restrict__ A,
                    const float* __restrict__ W,
                    const float* __restrict__ bias,
                    float* __restrict__ C,
                    int K) {
    extern __shared__ float smem[];
    float* As = smem;               // 128 x 66 floats
    float* Wf = smem + 128 * 66;    // 32 fragment-rows x 128 cols x float2

    const int tid  = threadIdx.x;
    const int wave = tid >> 5;
    const int lane = tid & 31;
    const int half = lane >> 4;     // 0: lanes 0-15, 1: lanes 16-31
    const int ml   = lane & 15;
    const int m0   = blockIdx.x * 128;

    v8f acc[8];
#pragma unroll
    for (int nt = 0; nt < 8; ++nt) acc[nt] = {};

    for (int kc = 0; kc < K; kc += 64) {
        __syncthreads();
        // stage A chunk: 128 rows x 64 cols (row-padded to 66)
        for (int i = tid; i < 128 * 64; i += 256) {
            const int r = i >> 6, c = i & 63;
            As[r * 66 + c] = A[(size_t)(m0 + r) * K + kc + c];
        }
        // stage W chunk fragment-major: for (kk,half,n) store the pair
        // { W[k0+half][n], W[k0+half+2][n] } contiguously.
        for (int i = tid; i < 32 * 128; i += 256) {
            const int kkh = i >> 7;           // 0..31 = kk*2 + half
            const int n   = i & 127;
            const int r0  = (kkh >> 1) * 4 + (kkh & 1);
            Wf[2 * i + 0] = W[(size_t)(kc + r0) * 128 + n];
            Wf[2 * i + 1] = W[(size_t)(kc + r0 + 2) * 128 + n];
        }
        __syncthreads();

        const float* Arow = &As[(wave * 16 + ml) * 66];
        for (int kk = 0; kk < 16; ++kk) {
            // A fragment (16x4 f32): lanes0-15 hold {K=k0,k0+1}, lanes16-31 {k0+2,k0+3}
            const v2f a = *reinterpret_cast<const v2f*>(&Arow[kk * 4 + half * 2]);
            const float* Wrow = &Wf[2 * ((kk * 2 + half) * 128)];
#pragma unroll
            for (int nt = 0; nt < 8; ++nt) {
                const v2f bf = *reinterpret_cast<const v2f*>(&Wrow[2 * (nt * 16 + ml)]);
                acc[nt] = __builtin_amdgcn_wmma_f32_16x16x4_f32(
                    false, a, false, bf, (short)0, acc[nt], false, false);
            }
        }
    }

    // epilogue: C/D layout — VGPR r holds M = r (lanes 0-15) / r+8 (lanes 16-31)
#pragma unroll
    for (int nt = 0; nt < 8; ++nt) {
        const int col = nt * 16 + ml;
        const float bv = bias[col];
#pragma unroll
        for (int r = 0; r < 8; ++r) {
            const int row = m0 + wave * 16 + half * 8 + r;
            C[(size_t)row * 128 + col] = acc[nt][r] + bv;
        }
    }
}

// ---------------------------------------------------------------------------
// Kernel 4: GroupNorm statistics. One block per (batch, group); reduce
// sum/sumsq over NF rows x 4 channels with float4 loads.
// ---------------------------------------------------------------------------
__global__ __launch_bounds__(256)
void gn_stats_kernel(const float* __restrict__ H, float* __restrict__ stats) {
    __shared__ float ssum[256];
    __shared__ float ssq[256];
    const int b = blockIdx.x >> 5;
    const int g = blockIdx.x & 31;

    float s = 0.f, q = 0.f;
    for (int n = threadIdx.x; n < NF_; n += 256) {
        const float4 v = *reinterpret_cast<const float4*>(
            &H[((size_t)(b * NF_ + n)) * OUT_CH_ + g * 4]);
        s += v.x + v.y + v.z + v.w;
        q = fmaf(v.x, v.x, q); q = fmaf(v.y, v.y, q);
        q = fmaf(v.z, v.z, q); q = fmaf(v.w, v.w, q);
    }
    ssum[threadIdx.x] = s;
    ssq[threadIdx.x]  = q;
    __syncthreads();
    for (int off = 128; off > 0; off >>= 1) {
        if (threadIdx.x < off) {
            ssum[threadIdx.x] += ssum[threadIdx.x + off];
            ssq[threadIdx.x]  += ssq[threadIdx.x + off];
        }
        __syncthreads();
    }
    if (threadIdx.x == 0) {
        const float inv  = 1.f / (float)(NF_ * 4);
        const float mean = ssum[0] * inv;
        const float var  = ssq[0] * inv - mean * mean;
        stats[(size_t)blockIdx.x * 2 + 0] = mean;
        stats[(size_t)blockIdx.x * 2 + 1] = rsqrtf(var + EPS_);
    }
}

// ---------------------------------------------------------------------------
// Kernel 5: fused GroupNorm apply + affine + ReLU (elementwise).
// ---------------------------------------------------------------------------
__global__ __launch_bounds__(256)
void gn_apply_relu_kernel(const float* __restrict__ H,
                          const float* __restrict__ gamma,
                          const float* __restrict__ beta,
                          const float* __restrict__ stats,
                          float* __restrict__ Y) {
    const size_t i = (size_t)blockIdx.x * 256 + threadIdx.x;  // over B*NF*128
    const int c   = (int)(i & 127);
    const int row = (int)(i >> 7);
    const int b   = row / NF_;
    const int g   = c >> 2;
    const float mean = stats[((size_t)b * 32 + g) * 2 + 0];
    const float istd = stats[((size_t)b * 32 + g) * 2 + 1];
    const float y = fmaf((H[i] - mean) * istd, gamma[c], beta[c]);
    Y[i] = fmaxf(y, 0.f);
}

// ---------------------------------------------------------------------------
extern "C" void kernel_launch(void* const* d_in, const int* in_sizes, int n_in,
                              void* d_out, int out_size, void* d_ws, size_t ws_size,
                              hipStream_t stream) {
    const float* xyz_coarse  = (const float*)d_in[0];
    const float* feat_coarse = (const float*)d_in[1];
    const float* xyz_fine    = (const float*)d_in[2];
    const float* feat_skip   = (const float*)d_in[3];
    const float* W1  = (const float*)d_in[4];
    const float* b1  = (const float*)d_in[5];
    const float* g1  = (const float*)d_in[6];
    const float* be1 = (const float*)d_in[7];
    const float* W2  = (const float*)d_in[8];
    const float* b2  = (const float*)d_in[9];
    const float* g2  = (const float*)d_in[10];
    const float* be2 = (const float*)d_in[11];
    float* out = (float*)d_out;

    char* ws = (char*)d_ws;
    size_t o = 0;
    int*   knn_idx = (int*)(ws + o);   o += (size_t)B_ * NF_ * 3 * sizeof(int);
    float* knn_w   = (float*)(ws + o); o += (size_t)B_ * NF_ * 3 * sizeof(float);
    float* feat    = (float*)(ws + o); o += (size_t)B_ * NF_ * IN_CH_ * sizeof(float);
    float* h1      = (float*)(ws + o); o += (size_t)B_ * NF_ * OUT_CH_ * sizeof(float);
    float* h2      = (float*)(ws + o); o += (size_t)B_ * NF_ * OUT_CH_ * sizeof(float);
    float* stats1  = (float*)(ws + o); o += (size_t)B_ * 32 * 2 * sizeof(float);
    float* stats2  = (float*)(ws + o);

    const int M = B_ * NF_;                       // 65536 rows
    const size_t gemm_smem = (128 * 66 + 32 * 128 * 2) * sizeof(float);  // ~65 KB dynamic LDS

    knn_kernel<<<M / 256, 256, 0, stream>>>(xyz_fine, xyz_coarse, knn_idx, knn_w);
    interp_concat_kernel<<<M, 128, 0, stream>>>(feat_coarse, feat_skip, knn_idx, knn_w, feat);

    gemm_bias_wmma<<<M / 128, 256, gemm_smem, stream>>>(feat, W1, b1, h1, IN_CH_);
    gn_stats_kernel<<<B_ * 32, 256, 0, stream>>>(h1, stats1);
    gn_apply_relu_kernel<<<(M * OUT_CH_) / 256, 256, 0, stream>>>(h1, g1, be1, stats1, h1);

    gemm_bias_wmma<<<M / 128, 256, gemm_smem, stream>>>(h1, W2, b2, h2, OUT_CH_);
    gn_stats_kernel<<<B_ * 32, 256, 0, stream>>>(h2, stats2);
    gn_apply_relu_kernel<<<(M * OUT_CH_) / 256, 256, 0, stream>>>(h2, g2, be2, stats2, out);
}